// MambaBlock_70076686402475
// MI455X (gfx1250) — compile-verified
//
#include <hip/hip_runtime.h>
#include <hip/hip_bf16.h>
#include <math.h>

// ---- problem constants (match reference) ----
#define D_MODEL 768
#define ED      1536
#define NST     16
#define DTR     48
#define KCONV   4
#define BATCH   16
#define HH      14
#define WIN     7
#define LSEQ    197            // H*H + 1
#define M_MAIN  (BATCH * LSEQ) // 3152 = 197*16
#define M_WIN   (4 * BATCH * WIN * WIN) // 3136 = 196*16
#define DBC_N   (DTR + 2 * NST) // 80

typedef float  v2f  __attribute__((ext_vector_type(2)));
typedef float  v8f  __attribute__((ext_vector_type(8)));

__device__ __forceinline__ float silu_f(float x) {
    return x / (1.0f + __expf(-x));
}
__device__ __forceinline__ float softplus_f(float x) {
    return (x > 20.0f) ? x : log1pf(__expf(x));
}

// =====================================================================
// fp32 WMMA GEMM:  C[M,N] = A[M,K] @ W[N,K]^T   (+ fused epilogue)
// Each wave computes a (16*TM) x (16*TN) macro-tile with
// V_WMMA_F32_16X16X4_F32, software-pipelined: fragment loads for step
// k+4 are issued before the WMMAs of step k, so matrix issue overlaps
// VMEM latency instead of waiting loadcnt==0 per WMMA.
// Ragged M edge handled by clamping A rows (duplicate compute, store
// skipped), so M only needs to be a multiple of 16.
// EPI: 0 = plain store (ldc = N)
//      1 = split: n < N/2 -> C0 (ldc N/2); n >= N/2 -> silu -> C1
//      2 = softplus(acc + bias[n]) -> C0 (ldc = N)
// =====================================================================
template <int TM, int TN, int EPI>
__global__ __launch_bounds__(256)
void gemm_wmma_f32(const float* __restrict__ A, const float* __restrict__ W,
                   float* __restrict__ C0, float* __restrict__ C1,
                   const float* __restrict__ bias,
                   int M, int N, int K, int lda, int ldb) {
    const int lane = threadIdx.x & 31;
    const int wid  = blockIdx.x * (blockDim.x >> 5) + (threadIdx.x >> 5);
    const int MT  = M >> 4;                 // 16-row tiles
    const int MTg = (MT + TM - 1) / TM;     // macro-tile groups in M
    const int NG  = N / (16 * TN);
    if (wid >= MTg * NG) return;

    const int mg = wid % MTg;
    const int ng = wid / MTg;
    const int m0 = mg * 16 * TM;
    const int n0 = ng * 16 * TN;

    const int r    = lane & 15;
    const int half = lane >> 4;

    const float* Ar[TM];
#pragma unroll
    for (int tm = 0; tm < TM; ++tm) {
        int row = m0 + tm * 16 + r;
        if (row >= M) row = M - 1;          // clamp ragged edge (store skipped)
        Ar[tm] = A + (size_t)row * lda + 2 * half;
    }
    const float* Br[TN];
#pragma unroll
    for (int tn = 0; tn < TN; ++tn)
        Br[tn] = W + (size_t)(n0 + tn * 16 + r) * ldb + 2 * half;

    v8f acc[TM][TN];
#pragma unroll
    for (int tm = 0; tm < TM; ++tm)
#pragma unroll
        for (int tn = 0; tn < TN; ++tn) acc[tm][tn] = (v8f){0,0,0,0,0,0,0,0};

    // prologue: fragments for k = 0
    v2f a_c[TM], b_c[TN];
#pragma unroll
    for (int tm = 0; tm < TM; ++tm) a_c[tm] = *(const v2f*)(Ar[tm]);
#pragma unroll
    for (int tn = 0; tn < TN; ++tn) b_c[tn] = *(const v2f*)(Br[tn]);

#pragma unroll 2
    for (int k = 0; k < K; k += 4) {
        // issue next-step loads first (clamped so they are unconditional)
        const int kn = (k + 4 < K) ? (k + 4) : k;
        v2f a_n[TM], b_n[TN];
#pragma unroll
        for (int tm = 0; tm < TM; ++tm) a_n[tm] = *(const v2f*)(Ar[tm] + kn);
#pragma unroll
        for (int tn = 0; tn < TN; ++tn) b_n[tn] = *(const v2f*)(Br[tn] + kn);

        // matrix ops consume last iteration's fragments
#pragma unroll
        for (int tm = 0; tm < TM; ++tm)
#pragma unroll
            for (int tn = 0; tn < TN; ++tn)
                acc[tm][tn] = __builtin_amdgcn_wmma_f32_16x16x4_f32(
                    false, a_c[tm], false, b_c[tn], (short)0, acc[tm][tn],
                    false, false);

#pragma unroll
        for (int tm = 0; tm < TM; ++tm) a_c[tm] = a_n[tm];
#pragma unroll
        for (int tn = 0; tn < TN; ++tn) b_c[tn] = b_n[tn];
    }

    const int halfN = N >> 1;
#pragma unroll
    for (int tm = 0; tm < TM; ++tm) {
        if (m0 + tm * 16 >= M) continue;    // ragged edge: no store
#pragma unroll
        for (int tn = 0; tn < TN; ++tn) {
            const int col = n0 + tn * 16 + r;
#pragma unroll
            for (int v = 0; v < 8; ++v) {
                const size_t row = (size_t)(m0 + tm * 16 + v + 8 * half);
                const float val = acc[tm][tn][v];
                if (EPI == 0) {
                    C0[row * (size_t)N + col] = val;
                } else if (EPI == 1) {
                    if (col < halfN) C0[row * (size_t)halfN + col] = val;
                    else C1[row * (size_t)halfN + (col - halfN)] = silu_f(val);
                } else { // EPI == 2
                    C0[row * (size_t)N + col] = softplus_f(val + bias[col]);
                }
            }
        }
    }
}

// =====================================================================
// Causal depthwise conv (K=4) + bias + silu.
// mode 0: identity sequence (branch 0)
// mode 1: spatially transposed sequence (branch 1):
//         x2(b, i=w*14+h, e) = xin(b, h*14+w, e); i==196 -> 196 (class tok)
// =====================================================================
__global__ __launch_bounds__(256)
void dwconv_silu(const float* __restrict__ xin, const float* __restrict__ w,
                 const float* __restrict__ bias, float* __restrict__ u, int mode) {
    size_t idx = (size_t)blockIdx.x * blockDim.x + threadIdx.x;
    if (idx >= (size_t)M_MAIN * ED) return;
    const int e = (int)(idx % ED);
    const int m = (int)(idx / ED);
    const int t = m % LSEQ;
    const int b = m / LSEQ;

    float acc = bias[e];
#pragma unroll
    for (int j = 0; j < KCONV; ++j) {
        const int tt = t - (KCONV - 1) + j;
        if (tt < 0) continue;
        int ts = tt;
        if (mode == 1 && tt < LSEQ - 1) {
            const int wv = tt / HH, hv = tt % HH;
            ts = hv * HH + wv;
        }
        acc += w[e * KCONV + j] * xin[((size_t)b * LSEQ + ts) * ED + e];
    }
    u[idx] = silu_f(acc);
}

// =====================================================================
// Window prep: uw(s=q*16+b, p, e) = silu(xin(b, t', e)*w2[e] + b2[e])
// =====================================================================
__global__ __launch_bounds__(256)
void win_prep(const float* __restrict__ xin, const float* __restrict__ w2,
              const float* __restrict__ b2, float* __restrict__ uw) {
    size_t idx = (size_t)blockIdx.x * blockDim.x + threadIdx.x;
    if (idx >= (size_t)M_WIN * ED) return;
    const int e = (int)(idx % ED);
    const int m = (int)(idx / ED);
    const int p = m % (WIN * WIN);
    const int s = m / (WIN * WIN);
    const int q = s / BATCH;
    const int b = s % BATCH;
    const int h = WIN * (q >> 1) + p / WIN;
    const int w = WIN * (q & 1) + p % WIN;
    const float x = xin[((size_t)b * LSEQ + h * HH + w) * ED + e] * w2[e] + b2[e];
    uw[idx] = silu_f(x);
}

// =====================================================================
// Selective scan: for each (seq s, channel e) recurrence over t with 16
// states. B/C (dbc[48:80]) staged through LDS per timestep.
//   h[n] = exp(delta*A[n])*h[n] + delta*u*B[n] ; y = sum h[n]*C[n]
// grid.x = S * (ED/256), block 256
// =====================================================================
__global__ __launch_bounds__(256)
void ssm_scan(const float* __restrict__ u, const float* __restrict__ delta,
              const float* __restrict__ dbc, const float* __restrict__ A_log,
              float* __restrict__ y, int S, int T) {
    __shared__ float sbc[2 * NST];
    const int chunks = ED / 256;
    const int s = blockIdx.x / chunks;
    const int e = (blockIdx.x % chunks) * 256 + threadIdx.x;
    if (s >= S) return;

    float An[NST];
#pragma unroll
    for (int n = 0; n < NST; ++n) An[n] = -__expf(A_log[(size_t)e * NST + n]);

    float h[NST];
#pragma unroll
    for (int n = 0; n < NST; ++n) h[n] = 0.0f;

    for (int t = 0; t < T; ++t) {
        __syncthreads();
        if (threadIdx.x < 2 * NST)
            sbc[threadIdx.x] = dbc[((size_t)s * T + t) * DBC_N + DTR + threadIdx.x];
        __syncthreads();

        const size_t off = ((size_t)s * T + t) * ED + e;
        const float d  = delta[off];
        const float du = d * u[off];
        float acc = 0.0f;
#pragma unroll
        for (int n = 0; n < NST; ++n) {
            h[n] = __expf(d * An[n]) * h[n] + du * sbc[n];
            acc += h[n] * sbc[NST + n];
        }
        y[off] = acc;
    }
}

// =====================================================================
// Combine the three branch outputs with permutations, multiply by z.
// =====================================================================
__global__ __launch_bounds__(256)
void combine(const float* __restrict__ y0, const float* __restrict__ y1,
             const float* __restrict__ yw, const float* __restrict__ z,
             float* __restrict__ out) {
    size_t idx = (size_t)blockIdx.x * blockDim.x + threadIdx.x;
    if (idx >= (size_t)M_MAIN * ED) return;
    const int e = (int)(idx % ED);
    const int m = (int)(idx / ED);
    const int t = m % LSEQ;
    const int b = m / LSEQ;

    float o;
    if (t == 0) {
        o = y0[((size_t)b * LSEQ + (LSEQ - 1)) * ED + e]
          + y1[((size_t)b * LSEQ + (LSEQ - 1)) * ED + e];
    } else {
        const int tp = t - 1;
        const int hh = tp / HH, ww = tp % HH;
        o = y0[((size_t)b * LSEQ + tp) * ED + e]
          + y1[((size_t)b * LSEQ + (ww * HH + hh)) * ED + e]; // undo transpose
    }
    if (t < LSEQ - 1) {
        const int h = t / HH, w = t % HH;
        const int q = ((h >= WIN) ? 2 : 0) + ((w >= WIN) ? 1 : 0);
        const int p = (h % WIN) * WIN + (w % WIN);
        o += yw[(((size_t)(q * BATCH + b)) * (WIN * WIN) + p) * ED + e];
    } else {
        o += y1[((size_t)b * LSEQ + (LSEQ - 1)) * ED + e]; // ct1
    }
    out[idx] = o * z[idx];
}

// =====================================================================
// host-side launch helpers
// =====================================================================
template <int TM, int TN, int EPI>
static void launch_gemm(const float* A, const float* W, float* C0, float* C1,
                        const float* bias, int M, int N, int K, int lda, int ldb,
                        hipStream_t s) {
    const int MTg   = ((M / 16) + TM - 1) / TM;
    const int waves = MTg * (N / (16 * TN));
    const int blocks = (waves + 7) / 8;
    gemm_wmma_f32<TM, TN, EPI><<<blocks, 256, 0, s>>>(A, W, C0, C1, bias,
                                                      M, N, K, lda, ldb);
}

extern "C" void kernel_launch(void* const* d_in, const int* in_sizes, int n_in,
                              void* d_out, int out_size, void* d_ws, size_t ws_size,
                              hipStream_t stream) {
    const float* x       = (const float*)d_in[0];   // (16,197,768)
    const float* W_in    = (const float*)d_in[1];   // (3072,768)
    const float* conv0_w = (const float*)d_in[2];   // (1536,1,4)
    const float* conv0_b = (const float*)d_in[3];
    const float* conv1_w = (const float*)d_in[4];
    const float* conv1_b = (const float*)d_in[5];
    const float* conv2_w = (const float*)d_in[6];   // (1536,1,1)
    const float* conv2_b = (const float*)d_in[7];
    const float* xproj0  = (const float*)d_in[8];   // (80,1536)
    const float* xproj1  = (const float*)d_in[9];
    const float* xproj2  = (const float*)d_in[10];
    const float* dt_w    = (const float*)d_in[11];  // (1536,48)
    const float* dt_b    = (const float*)d_in[12];  // (1536,)
    const float* A_log   = (const float*)d_in[13];  // (1536,16)
    const float* W_out   = (const float*)d_in[15];  // (768,1536)
    float* out = (float*)d_out;                     // (16,197,768)

    // ---- workspace layout (floats) ----
    const size_t SZ = (size_t)M_MAIN * ED;          // 4,841,472
    float* ws   = (float*)d_ws;
    float* xin  = ws + 0 * SZ;
    float* z    = ws + 1 * SZ;
    float* y0   = ws + 2 * SZ;
    float* y1   = ws + 3 * SZ;
    float* yw   = ws + 4 * SZ;
    float* ubuf = ws + 5 * SZ;   // reused: u0 -> u1 -> uw -> combined output
    float* dbuf = ws + 6 * SZ;   // delta scratch
    float* dbc  = ws + 7 * SZ;   // (M, 80) scratch

    const int EB = (int)(((size_t)M_MAIN * ED + 255) / 256);
    const int WB = (int)(((size_t)M_WIN  * ED + 255) / 256);

    // 1) xz = x @ W_in^T ; split into xin / silu(z)
    launch_gemm<2, 4, 1>(x, W_in, xin, z, nullptr, M_MAIN, 2 * ED, D_MODEL,
                         D_MODEL, D_MODEL, stream);

    // 2) branch 0: dwconv -> dbc -> delta -> scan -> y0
    dwconv_silu<<<EB, 256, 0, stream>>>(xin, conv0_w, conv0_b, ubuf, 0);
    launch_gemm<2, 5, 0>(ubuf, xproj0, dbc, nullptr, nullptr, M_MAIN, DBC_N, ED,
                         ED, ED, stream);
    launch_gemm<2, 4, 2>(dbc, dt_w, dbuf, nullptr, dt_b, M_MAIN, ED, DTR,
                         DBC_N, DTR, stream);
    ssm_scan<<<BATCH * (ED / 256), 256, 0, stream>>>(ubuf, dbuf, dbc, A_log,
                                                     y0, BATCH, LSEQ);

    // 3) branch 1: transposed sequence
    dwconv_silu<<<EB, 256, 0, stream>>>(xin, conv1_w, conv1_b, ubuf, 1);
    launch_gemm<2, 5, 0>(ubuf, xproj1, dbc, nullptr, nullptr, M_MAIN, DBC_N, ED,
                         ED, ED, stream);
    launch_gemm<2, 4, 2>(dbc, dt_w, dbuf, nullptr, dt_b, M_MAIN, ED, DTR,
                         DBC_N, DTR, stream);
    ssm_scan<<<BATCH * (ED / 256), 256, 0, stream>>>(ubuf, dbuf, dbc, A_log,
                                                     y1, BATCH, LSEQ);

    // 4) window branch: 64 sequences of length 49
    win_prep<<<WB, 256, 0, stream>>>(xin, conv2_w, conv2_b, ubuf);
    launch_gemm<2, 5, 0>(ubuf, xproj2, dbc, nullptr, nullptr, M_WIN, DBC_N, ED,
                         ED, ED, stream);
    launch_gemm<2, 4, 2>(dbc, dt_w, dbuf, nullptr, dt_b, M_WIN, ED, DTR,
                         DBC_N, DTR, stream);
    ssm_scan<<<4 * BATCH * (ED / 256), 256, 0, stream>>>(ubuf, dbuf, dbc, A_log,
                                                         yw, 4 * BATCH, WIN * WIN);

    // 5) combine branches * z  -> ubuf (now free)
    combine<<<EB, 256, 0, stream>>>(y0, y1, yw, z, ubuf);

    // 6) out = combined @ W_out^T
    launch_gemm<2, 4, 0>(ubuf, W_out, out, nullptr, nullptr, M_MAIN, D_MODEL, ED,
                         ED, ED, stream);
}